// Net_58153857187993
// MI455X (gfx1250) — compile-verified
//
#include <hip/hip_runtime.h>
#include <math.h>

typedef float v2f __attribute__((ext_vector_type(2)));
typedef float v8f __attribute__((ext_vector_type(8)));

#define NEG_SLOPE 0.2f

__device__ __forceinline__ float leaky(float v) { return v >= 0.f ? v : NEG_SLOPE * v; }

// sign-aware float atomic max (init must be -inf)
__device__ __forceinline__ void atomicMaxFloat(float* addr, float val) {
    if (val >= 0.f) atomicMax((int*)addr, __float_as_int(val));
    else            atomicMin((unsigned int*)addr, __float_as_uint(val));
}

// ---------------- fp32 WMMA GEMM: C[M,Nc] = A[M,K] * B[K,Nc] ----------------
// One wave32 per 16x16 output tile, V_WMMA_F32_16X16X4_F32, K stepped by 4.
// A-frag (16x4): lanes 0-15 hold K=k+0,k+1 ; lanes 16-31 hold K=k+2,k+3 (ISA 7.12.2)
// B-frag (4x16): same K split across lane halves, N = lane&15
// C/D (16x16):   VGPR r -> M = r + 8*(lane>=16), N = lane&15
__global__ void gemm_wmma_f32(const float* __restrict__ A, const float* __restrict__ B,
                              float* __restrict__ C, int M, int K, int Nc) {
    const int lane = threadIdx.x & 31;
    const int wave = threadIdx.x >> 5;
    const int tilesPerRow = Nc >> 4;
    const int nTiles = (M >> 4) * tilesPerRow;
    const int tile = blockIdx.x * (blockDim.x >> 5) + wave;
    if (tile >= nTiles) return;
    const int rowTile = (tile / tilesPerRow) << 4;
    const int colTile = (tile % tilesPerRow) << 4;
    const int mn   = lane & 15;   // A row within tile / B,C col within tile
    const int half = lane >> 4;   // K sub-block selector
    v8f c = {};
    const float* arow = A + (size_t)(rowTile + mn) * K;
    const float* bcol = B + colTile + mn;
    for (int k = 0; k < K; k += 4) {
        const int ka = k + 2 * half;
        v2f a = *(const v2f*)(arow + ka);          // 8B aligned (ka even)
        v2f b;
        b.x = bcol[(size_t)ka * Nc];
        b.y = bcol[(size_t)(ka + 1) * Nc];
        c = __builtin_amdgcn_wmma_f32_16x16x4_f32(false, a, false, b,
                                                  (short)0, c, false, false);
    }
    float* crow = C + (size_t)(rowTile + half * 8) * Nc + colTile + mn;
#pragma unroll
    for (int r = 0; r < 8; ++r) crow[(size_t)r * Nc] = c[r];
}

// ---------------- misc elementwise kernels ----------------
__global__ void fill_f32(float* __restrict__ p, float v, int n) {
    int i = blockIdx.x * blockDim.x + threadIdx.x;
    if (i < n) p[i] = v;
}

// per-(node,head) attention logits: als/ald[n*H+h] = sum_c h[n,h,c]*a_{src,dst}[h,c]
__global__ void att_logits(const float* __restrict__ h, const float* __restrict__ a_src,
                           const float* __restrict__ a_dst, float* __restrict__ als,
                           float* __restrict__ ald, int N_, int H, int Ch) {
    int idx = blockIdx.x * blockDim.x + threadIdx.x;
    if (idx >= N_ * H) return;
    int n = idx / H, hh = idx % H;
    const float* hp = h + (size_t)n * H * Ch + hh * Ch;
    const float* as = a_src + hh * Ch;
    const float* ad = a_dst + hh * Ch;
    float s = 0.f, d = 0.f;
    for (int c = 0; c < Ch; ++c) { float v = hp[c]; s += v * as[c]; d += v * ad[c]; }
    als[idx] = s; ald[idx] = d;
}

// pass 1: segment max of leaky_relu(als[src]+ald[dst]) over dst
__global__ void edge_max(const int* __restrict__ src, const int* __restrict__ dst, int E, int N_,
                         const float* __restrict__ als, const float* __restrict__ ald,
                         float* __restrict__ m, int H) {
    int e = blockIdx.x * blockDim.x + threadIdx.x;
    if (e >= E + N_) return;
    int s = (e < E) ? src[e] : (e - E);
    int d = (e < E) ? dst[e] : (e - E);
    for (int hh = 0; hh < H; ++hh) {
        float v = leaky(als[(size_t)s * H + hh] + ald[(size_t)d * H + hh]);
        atomicMaxFloat(&m[(size_t)d * H + hh], v);
    }
}

// pass 2: denom[dst] += exp(e - m[dst])
__global__ void edge_den(const int* __restrict__ src, const int* __restrict__ dst, int E, int N_,
                         const float* __restrict__ als, const float* __restrict__ ald,
                         const float* __restrict__ m, float* __restrict__ den, int H) {
    int e = blockIdx.x * blockDim.x + threadIdx.x;
    if (e >= E + N_) return;
    int s = (e < E) ? src[e] : (e - E);
    int d = (e < E) ? dst[e] : (e - E);
    for (int hh = 0; hh < H; ++hh) {
        float v  = leaky(als[(size_t)s * H + hh] + ald[(size_t)d * H + hh]);
        float ex = expf(v - m[(size_t)d * H + hh]);
        atomicAdd(&den[(size_t)d * H + hh], ex);
    }
}

// pass 3: out[dst,:] += alpha * h[src,:]; one wave32 per edge, F = H*Ch channels
__global__ void edge_aggr(const int* __restrict__ src, const int* __restrict__ dst, int E, int N_,
                          const float* __restrict__ als, const float* __restrict__ ald,
                          const float* __restrict__ m, const float* __restrict__ den,
                          const float* __restrict__ h, float* __restrict__ out, int H, int Ch) {
    int gid = blockIdx.x * blockDim.x + threadIdx.x;
    int lane = gid & 31;
    int e = gid >> 5;
    if (e >= E + N_) return;
    int s = (e < E) ? src[e] : (e - E);
    int d = (e < E) ? dst[e] : (e - E);
    const int F = H * Ch;
    const int cpl = F >> 5;  // channels per lane (4 for F=128, 1 for F=32)
    for (int j = 0; j < cpl; ++j) {
        int c  = lane * cpl + j;
        int hh = c / Ch;
        float v     = leaky(als[(size_t)s * H + hh] + ald[(size_t)d * H + hh]);
        float alpha = expf(v - m[(size_t)d * H + hh]) /
                      (den[(size_t)d * H + hh] + 1e-16f);
        atomicAdd(&out[(size_t)d * F + c], h[(size_t)s * F + c] * alpha);
    }
}

// x = elu(x + bias[i % F])
__global__ void elu_bias(float* __restrict__ x, const float* __restrict__ b, int n, int F) {
    int i = blockIdx.x * blockDim.x + threadIdx.x;
    if (i >= n) return;
    float v = x[i] + b[i % F];
    x[i] = v > 0.f ? v : expm1f(v);
}

// one wave32 per node: out[n,:] = log_softmax(out[n,:] + b2)
__global__ void logsoftmax32(float* __restrict__ out, const float* __restrict__ b2, int N_) {
    int gid = blockIdx.x * blockDim.x + threadIdx.x;
    int lane = gid & 31;
    int n = gid >> 5;
    if (n >= N_) return;
    float v = out[(size_t)n * 32 + lane] + b2[lane];
    float mx = v;
    for (int off = 16; off; off >>= 1) mx = fmaxf(mx, __shfl_xor(mx, off, 32));
    float ex = expf(v - mx);
    float sm = ex;
    for (int off = 16; off; off >>= 1) sm += __shfl_xor(sm, off, 32);
    out[(size_t)n * 32 + lane] = v - mx - logf(sm);
}

extern "C" void kernel_launch(void* const* d_in, const int* in_sizes, int n_in,
                              void* d_out, int out_size, void* d_ws, size_t ws_size,
                              hipStream_t stream) {
    (void)n_in; (void)out_size; (void)ws_size;
    const float* x    = (const float*)d_in[0];
    const int*   ei   = (const int*)d_in[1];
    const float* W1   = (const float*)d_in[2];
    const float* a1s  = (const float*)d_in[3];
    const float* a1d  = (const float*)d_in[4];
    const float* b1   = (const float*)d_in[5];
    const float* W2   = (const float*)d_in[6];
    const float* a2s  = (const float*)d_in[7];
    const float* a2d  = (const float*)d_in[8];
    const float* b2   = (const float*)d_in[9];
    float* out = (float*)d_out;

    const int N_ = in_sizes[0] / 128;   // 50000
    const int E  = in_sizes[1] / 2;     // 800000
    const int* src = ei;                // edge_index[0]
    const int* dst = ei + E;            // edge_index[1]
    const int ET = E + N_;              // edges + self loops

    // workspace layout (floats)
    float* ws = (float*)d_ws;
    float* h1   = ws; ws += (size_t)N_ * 128;  // x @ W1
    float* o1   = ws; ws += (size_t)N_ * 128;  // layer-1 aggregate -> elu output
    float* al1s = ws; ws += (size_t)N_ * 8;
    float* al1d = ws; ws += (size_t)N_ * 8;
    float* m1   = ws; ws += (size_t)N_ * 8;
    float* dn1  = ws; ws += (size_t)N_ * 8;
    float* h2   = ws; ws += (size_t)N_ * 32;   // elu(h1) @ W2
    float* al2s = ws; ws += (size_t)N_;
    float* al2d = ws; ws += (size_t)N_;
    float* m2   = ws; ws += (size_t)N_;
    float* dn2  = ws; ws += (size_t)N_;

    const int TB = 256;
    const float NEG_INF = -__builtin_huge_valf();

    // ---- layer 1 ----
    {
        int tiles = (N_ / 16) * (128 / 16);
        gemm_wmma_f32<<<(tiles + 3) / 4, 128, 0, stream>>>(x, W1, h1, N_, 128, 128);
    }
    att_logits<<<(N_ * 8 + TB - 1) / TB, TB, 0, stream>>>(h1, a1s, a1d, al1s, al1d, N_, 8, 16);
    fill_f32<<<(N_ * 8 + TB - 1) / TB, TB, 0, stream>>>(m1, NEG_INF, N_ * 8);
    fill_f32<<<(N_ * 8 + TB - 1) / TB, TB, 0, stream>>>(dn1, 0.f, N_ * 8);
    fill_f32<<<(N_ * 128 + TB - 1) / TB, TB, 0, stream>>>(o1, 0.f, N_ * 128);
    edge_max<<<(ET + TB - 1) / TB, TB, 0, stream>>>(src, dst, E, N_, al1s, al1d, m1, 8);
    edge_den<<<(ET + TB - 1) / TB, TB, 0, stream>>>(src, dst, E, N_, al1s, al1d, m1, dn1, 8);
    edge_aggr<<<((size_t)ET * 32 + TB - 1) / TB, TB, 0, stream>>>(src, dst, E, N_, al1s, al1d,
                                                                  m1, dn1, h1, o1, 8, 16);
    elu_bias<<<(N_ * 128 + TB - 1) / TB, TB, 0, stream>>>(o1, b1, N_ * 128, 128);

    // ---- layer 2 ----
    {
        int tiles = (N_ / 16) * (32 / 16);
        gemm_wmma_f32<<<(tiles + 3) / 4, 128, 0, stream>>>(o1, W2, h2, N_, 128, 32);
    }
    att_logits<<<(N_ + TB - 1) / TB, TB, 0, stream>>>(h2, a2s, a2d, al2s, al2d, N_, 1, 32);
    fill_f32<<<(N_ + TB - 1) / TB, TB, 0, stream>>>(m2, NEG_INF, N_);
    fill_f32<<<(N_ + TB - 1) / TB, TB, 0, stream>>>(dn2, 0.f, N_);
    fill_f32<<<(N_ * 32 + TB - 1) / TB, TB, 0, stream>>>(out, 0.f, N_ * 32);
    edge_max<<<(ET + TB - 1) / TB, TB, 0, stream>>>(src, dst, E, N_, al2s, al2d, m2, 1);
    edge_den<<<(ET + TB - 1) / TB, TB, 0, stream>>>(src, dst, E, N_, al2s, al2d, m2, dn2, 1);
    edge_aggr<<<((size_t)ET * 32 + TB - 1) / TB, TB, 0, stream>>>(src, dst, E, N_, al2s, al2d,
                                                                  m2, dn2, h2, out, 1, 32);
    logsoftmax32<<<((size_t)N_ * 32 + TB - 1) / TB, TB, 0, stream>>>(out, b2, N_);
}